// SelfIntersections_40218073760193
// MI455X (gfx1250) — compile-verified
//
#include <hip/hip_runtime.h>
#include <hip/hip_bf16.h>

#define NTRI 2048
#define WAVES_PER_BLOCK 4

typedef __attribute__((ext_vector_type(2))) float v2f;
typedef __attribute__((ext_vector_type(8))) float v8f;

struct F3 { float x, y, z; };
struct F2 { float x, y; };

__device__ __forceinline__ F3 sub3(F3 a, F3 b) { return {a.x - b.x, a.y - b.y, a.z - b.z}; }
__device__ __forceinline__ F3 cross3(F3 a, F3 b) {
    return {a.y * b.z - a.z * b.y, a.z * b.x - a.x * b.z, a.x * b.y - a.y * b.x};
}
__device__ __forceinline__ float dot3(F3 a, F3 b) { return a.x * b.x + a.y * b.y + a.z * b.z; }
__device__ __forceinline__ F3 sel3(bool c, F3 a, F3 b) {
    return {c ? a.x : b.x, c ? a.y : b.y, c ? a.z : b.z};
}

__device__ __forceinline__ float orient2(F2 p, F2 q, F2 r) {
    return (p.x - r.x) * (q.y - r.y) - (p.y - r.y) * (q.x - r.x);
}

__device__ __forceinline__ bool cmm(F3 p1, F3 q1, F3 r1, F3 p2, F3 q2, F3 r2) {
    F3 n = cross3(sub3(p2, q1), sub3(p1, q1));
    bool c1 = dot3(sub3(q2, q1), n) > 0.f;
    F3 m = cross3(sub3(p2, p1), sub3(r1, p1));
    bool c2 = dot3(sub3(r2, p1), m) > 0.f;
    return !(c1 | c2);
}

__device__ bool itv2(F2 p1, F2 q1, F2 r1, F2 p2, F2 q2, F2 r2) {
    bool A = orient2(r2, p2, q1) >= 0.f;
    bool B = orient2(r2, q2, q1) <= 0.f;
    bool C = orient2(p1, p2, q1) > 0.f;
    bool D = orient2(p1, q2, q1) <= 0.f;
    bool E = orient2(p1, p2, r1) >= 0.f;
    bool F = orient2(q1, r1, p2) >= 0.f;
    bool G = orient2(r2, q2, r1) <= 0.f;
    bool H = orient2(q1, r1, r2) >= 0.f;
    bool I = orient2(r2, p2, r1) >= 0.f;
    bool J = orient2(q1, r1, q2) >= 0.f;
    bool K = orient2(r2, r1, q2) >= 0.f;
    bool tA  = B ? (C ? D : (E & F)) : (D & G & H);
    bool tNA = I & (H ? E : (J & K));
    return A ? tA : tNA;
}

__device__ bool ite2(F2 p1, F2 q1, F2 r1, F2 p2, F2 q2, F2 r2) {
    bool A = orient2(r2, p2, q1) >= 0.f;
    bool B = orient2(p1, p2, q1) >= 0.f;
    bool C = orient2(p1, q1, r2) >= 0.f;
    bool D = orient2(q1, r1, p2) >= 0.f;
    bool E = orient2(r1, p1, p2) >= 0.f;
    bool F = orient2(r2, p2, r1) >= 0.f;
    bool G = orient2(p1, p2, r1) >= 0.f;
    bool H = orient2(p1, r1, r2) >= 0.f;
    bool I = orient2(q1, r1, r2) >= 0.f;
    return A ? (B ? C : (D & E)) : (F & G & (H | I));
}

__device__ bool ccw2(F2 p1, F2 q1, F2 r1, F2 p2, F2 q2, F2 r2) {
    bool a = orient2(p2, q2, p1) >= 0.f;
    bool b = orient2(q2, r2, p1) >= 0.f;
    bool c = orient2(r2, p2, p1) >= 0.f;
    if (a) {
        if (b) return c ? true : ite2(p1, q1, r1, p2, q2, r2);
        else   return c ? ite2(p1, q1, r1, r2, p2, q2) : itv2(p1, q1, r1, p2, q2, r2);
    } else {
        if (b) return c ? ite2(p1, q1, r1, q2, r2, p2) : itv2(p1, q1, r1, q2, r2, p2);
        else   return itv2(p1, q1, r1, r2, p2, q2);
    }
}

__device__ bool tt2d(F2 p1, F2 q1, F2 r1, F2 p2, F2 q2, F2 r2) {
    bool s1 = orient2(p1, q1, r1) < 0.f;
    bool s2 = orient2(p2, q2, r2) < 0.f;
    F2 q1n = s1 ? r1 : q1, r1n = s1 ? q1 : r1;
    F2 q2n = s2 ? r2 : q2, r2n = s2 ? q2 : r2;
    return ccw2(p1, q1n, r1n, p2, q2n, r2n);
}

// Requires exact zero plane distances -> essentially never taken on random
// float input. Keep it out of line so the hot 8x-unrolled loop stays small.
__device__ __attribute__((noinline))
bool coplanar_tt(F3 p1, F3 q1, F3 r1, F3 p2, F3 q2, F3 r2, F3 n1) {
    float nx = fabsf(n1.x), ny = fabsf(n1.y), nz = fabsf(n1.z);
    bool cx = (nx > nz) & (nx >= ny);
    bool cy = (ny > nz) & (ny >= nx);
    auto proj = [&](F3 v) -> F2 {
        F2 r;
        r.x = cx ? v.z : v.x;
        r.y = cx ? v.y : (cy ? v.z : v.y);
        return r;
    };
    return tt2d(proj(q1), proj(p1), proj(r1), proj(q2), proj(p2), proj(r2));
}

__device__ __forceinline__
bool pairTest(const float* __restrict__ t1, const float* __restrict__ t2,
              float dp1, float dq1, float dr1,
              float dp2, float dq2, float dr2) {
    bool early1 = (dp1 * dq1 > 0.f) & (dp1 * dr1 > 0.f);
    bool early2 = (dp2 * dq2 > 0.f) & (dp2 * dr2 > 0.f);
    if (early1 | early2) return false;

    F3 p1{t1[0], t1[1], t1[2]}, q1{t1[3], t1[4], t1[5]}, r1{t1[6], t1[7], t1[8]};
    F3 p2{t2[0], t2[1], t2[2]}, q2{t2[3], t2[4], t2[5]}, r2{t2[6], t2[7], t2[8]};
    F3 n1 = cross3(sub3(q1, p1), sub3(r1, p1));

    // --- tri1 permutation: pure mask arithmetic (no control flow) -----
    // triA=(p,q,r)  triB=(q,r,p)  triC=(r,p,q)
    bool gp = dp1 > 0.f, gn = dp1 < 0.f;
    bool qP = dq1 > 0.f, qN = dq1 < 0.f;
    bool rP = dr1 > 0.f, rN = dr1 < 0.f;
    bool z  = (!gp) & (!gn);
    bool q0 = (!qP) & (!qN);
    bool rGE = !rN;          // dr1 >= 0
    bool rNZ = rP | rN;      // dr1 != 0

    bool isC = (gp & qP) | (gn & qN) | (z & q0 & rNZ);
    bool isB = (gp & (!qP) & rP) | (gn & (!qN) & rN)
             | (z & qN & rGE)    | (z & qP & (!rP));
    bool sw  = (gp & (qP | rP))  | (gn & (!(qN | rN)))
             | (z & qN & rGE)    | (z & qP & rP) | (z & q0 & rN);

    F3 P1 = sel3(isC, r1, sel3(isB, q1, p1));
    F3 Q1 = sel3(isC, p1, sel3(isB, r1, q1));
    F3 R1 = sel3(isC, q1, sel3(isB, p1, r1));

    F3 Q2 = sel3(sw, r2, q2);
    F3 R2 = sel3(sw, q2, r2);
    float DQ2 = sw ? dr2 : dq2;
    float DR2 = sw ? dq2 : dr2;

    // --- tt3d flattened: every live branch is cmm() with permuted args
    bool pp = dp2 > 0.f, pn = dp2 < 0.f;
    bool aP = DQ2 > 0.f, aN = DQ2 < 0.f;
    bool bP = DR2 > 0.f, bN = DR2 < 0.f;
    bool z2 = (!pp) & (!pn);
    bool a0 = (!aP) & (!aN);
    bool b0 = (!bP) & (!bN);

    bool ci1 = pp & aP;
    bool ci3 = (pp & (!aP) & (!bP)) | (z2 & aN & (!bP));
    bool ci4 = (pn & (aN | bN))     | (z2 & a0 & bP);
    bool ci5 = (pn & (!aN) & (!bN)) | (z2 & aP & bP);
    bool ci6 = z2 & aN & bP;
    bool ci7 = z2 & aP & (!bP);
    bool ci8 = z2 & a0 & bN;
    bool cop = z2 & a0 & b0;

    if (cop)
        return coplanar_tt(P1, Q1, R1, p2, Q2, R2, n1);

    // sel1: 0 -> (Q1,R1), 1 -> (R1,Q1), 2 -> (R1,Q2)  for cmm args 2,3
    // sel2: 0 -> (p2,Q2,R2), 1 -> (R2,p2,Q2), 2 -> (Q2,R2,p2) for args 4,5,6
    bool s1_0 = ci3 | ci4 | ci7;
    bool s1_2 = ci6 | ci8;
    bool s2_0 = ci3 | ci5 | ci6;
    bool s2_1 = ci1 | ci4 | ci8;

    F3 B1 = sel3(s1_0, Q1, R1);
    F3 C1 = sel3(s1_0, R1, sel3(s1_2, Q2, Q1));
    F3 A2 = sel3(s2_0, p2, sel3(s2_1, R2, Q2));
    F3 B2 = sel3(s2_0, Q2, sel3(s2_1, p2, R2));
    F3 C2 = sel3(s2_0, R2, sel3(s2_1, Q2, p2));
    return cmm(P1, B1, C1, A2, B2, C2);
}

__global__ __launch_bounds__(128) void SelfIntersections_kernel(
        const float* __restrict__ tris, float* __restrict__ out) {
    // rows shared by all 4 waves in the block (same tileRow); cols per-wave
    __shared__ float sRowT[16][9];
    __shared__ float sRowN[16][4];
    __shared__ float sColT[WAVES_PER_BLOCK][16][9];
    __shared__ float sColN[WAVES_PER_BLOCK][16][4];

    const int wave = threadIdx.x >> 5;
    const int lane = threadIdx.x & 31;
    const int tileRow = blockIdx.y;
    const int tileCol = blockIdx.x * WAVES_PER_BLOCK + wave;
    const int i0 = tileRow * 16;
    const int j0 = tileCol * 16;
    const bool upper = lane >= 16;
    const int  m = lane & 15;

    // entire tile at or below the diagonal (all j <= i) -> pure zero store
    const bool tileDead = (j0 + 15) <= i0;

    // wave 0, lanes 0..15 stage the 16 shared row triangles (unconditionally:
    // wave 0's own tile may be dead while sibling waves still need the rows)
    if (wave == 0 && !upper) {
        const float* src = tris + (size_t)(i0 + m) * 9;
        float t[9];
        #pragma unroll
        for (int k = 0; k < 9; ++k) t[k] = src[k];
        F3 P{t[0], t[1], t[2]}, Q{t[3], t[4], t[5]}, R{t[6], t[7], t[8]};
        F3 n = cross3(sub3(Q, P), sub3(R, P));   // n1 = cross(q1-p1, r1-p1)
        float d = -dot3(R, n);                   // dp2 = p2.n1 - r1.n1
        #pragma unroll
        for (int k = 0; k < 9; ++k) sRowT[m][k] = t[k];
        sRowN[m][0] = n.x; sRowN[m][1] = n.y; sRowN[m][2] = n.z; sRowN[m][3] = d;
    }
    // each wave's lanes 16..31 stage its 16 column triangles
    if (upper && !tileDead) {
        const float* src = tris + (size_t)(j0 + m) * 9;
        float t[9];
        #pragma unroll
        for (int k = 0; k < 9; ++k) t[k] = src[k];
        F3 P{t[0], t[1], t[2]}, Q{t[3], t[4], t[5]}, R{t[6], t[7], t[8]};
        F3 n = cross3(sub3(P, R), sub3(Q, R));   // n2 = cross(p2-r2, q2-r2)
        float d = -dot3(R, n);                   // dp1 = p1.n2 - r2.n2
        #pragma unroll
        for (int k = 0; k < 9; ++k) sColT[wave][m][k] = t[k];
        sColN[wave][m][0] = n.x; sColN[wave][m][1] = n.y;
        sColN[wave][m][2] = n.z; sColN[wave][m][3] = d;
    }
    __syncthreads();

    if (tileDead) {
        #pragma unroll
        for (int v = 0; v < 8; ++v) {
            int gi = i0 + v + (upper ? 8 : 0);
            int gj = j0 + m;
            out[(size_t)gi * NTRI + gj] = 0.0f;
        }
        return;
    }

    // Assemble A (16x4 f32: lanes 0-15 hold K=0,1; lanes 16-31 hold K=2,3)
    // and B (4x16 f32: same K split, N = lane%16) fragments.
    const float* rT = sRowT[m];
    const float* rN = sRowN[m];
    const float* cT = sColT[wave][m];
    const float* cN = sColN[wave][m];

    v2f a_p, a_q, a_r, a_n, b_n, b_p, b_q, b_r;
    if (!upper) {
        a_p = {rT[0], rT[1]};  a_q = {rT[3], rT[4]};  a_r = {rT[6], rT[7]};
        a_n = {rN[0], rN[1]};
        b_n = {cN[0], cN[1]};
        b_p = {cT[0], cT[1]};  b_q = {cT[3], cT[4]};  b_r = {cT[6], cT[7]};
    } else {
        a_p = {rT[2], 1.0f};   a_q = {rT[5], 1.0f};   a_r = {rT[8], 1.0f};
        a_n = {rN[2], rN[3]};
        b_n = {cN[2], cN[3]};
        b_p = {cT[2], 1.0f};   b_q = {cT[5], 1.0f};   b_r = {cT[8], 1.0f};
    }

    v8f zero = {};
    // D[i][j] = A_row(i) . B_col(j): six signed-distance 16x16 tiles
    v8f Dp1 = __builtin_amdgcn_wmma_f32_16x16x4_f32(false, a_p, false, b_n, (short)0, zero, false, false);
    v8f Dq1 = __builtin_amdgcn_wmma_f32_16x16x4_f32(false, a_q, false, b_n, (short)0, zero, false, false);
    v8f Dr1 = __builtin_amdgcn_wmma_f32_16x16x4_f32(false, a_r, false, b_n, (short)0, zero, false, false);
    v8f Dp2 = __builtin_amdgcn_wmma_f32_16x16x4_f32(false, a_n, false, b_p, (short)0, zero, false, false);
    v8f Dq2 = __builtin_amdgcn_wmma_f32_16x16x4_f32(false, a_n, false, b_q, (short)0, zero, false, false);
    v8f Dr2 = __builtin_amdgcn_wmma_f32_16x16x4_f32(false, a_n, false, b_r, (short)0, zero, false, false);

    // C/D layout: VGPR v -> M = v + (lane>=16 ? 8 : 0), N = lane%16
    #pragma unroll
    for (int v = 0; v < 8; ++v) {
        int il = v + (upper ? 8 : 0);
        int jl = m;
        int gi = i0 + il;
        int gj = j0 + jl;
        bool res = false;
        if (gj > gi) {
            res = pairTest(sRowT[il], sColT[wave][jl],
                           Dp1[v], Dq1[v], Dr1[v],
                           Dp2[v], Dq2[v], Dr2[v]);
        }
        out[(size_t)gi * NTRI + gj] = res ? 1.0f : 0.0f;
    }
}

extern "C" void kernel_launch(void* const* d_in, const int* in_sizes, int n_in,
                              void* d_out, int out_size, void* d_ws, size_t ws_size,
                              hipStream_t stream) {
    const float* tris = (const float*)d_in[0];
    float* out = (float*)d_out;
    dim3 block(32 * WAVES_PER_BLOCK);
    dim3 grid(NTRI / 16 / WAVES_PER_BLOCK, NTRI / 16);
    SelfIntersections_kernel<<<grid, block, 0, stream>>>(tris, out);
}